// ADN_SpGAT_26182120636487
// MI455X (gfx1250) — compile-verified
//
#include <hip/hip_runtime.h>
#include <hip/hip_bf16.h>

#define NNODES 50000
#define NEDGES 640000
#define NFEAT  128
#define NHID   128
#define ALPHA  0.2f

typedef __attribute__((ext_vector_type(2))) float v2f;
typedef __attribute__((ext_vector_type(8))) float v8f;

// ---------------------------------------------------------------------------
// Kernel 1: h = x @ W   (50000x128 @ 128x128, f32) via V_WMMA_F32_16X16X4_F32
// Block = 256 threads = 8 waves. Each block -> 16 rows x 128 cols of h.
// Wave w computes the 16x16 tile at cols [16w, 16w+16). K-loop steps by 4.
// A fragment (16x4 f32, 2 VGPR): lane L -> row m=L&15; lanes 0-15 hold K={k,k+1},
// lanes 16-31 hold K={k+2,k+3}  => contiguous float2 from row-major x.
// B fragment (4x16 f32, 2 VGPR): lane L -> col n=L&15; VGPR0/1 hold rows
// K = k+2*(L>>4) and k+2*(L>>4)+1 of W.
// C/D (16x16 f32, 8 VGPR): lane L -> col n=L&15, VGPR r -> row r + 8*(L>>4).
// ---------------------------------------------------------------------------
__global__ __launch_bounds__(256) void gat_gemm_wmma(
    const float* __restrict__ x, const float* __restrict__ W,
    float* __restrict__ h) {
  const int lane = threadIdx.x & 31;
  const int wave = threadIdx.x >> 5;           // 0..7 -> column tile
  const int row0 = blockIdx.x << 4;            // 16 rows per block (3125 blocks)
  const int col0 = wave << 4;
  const int m    = lane & 15;
  const int hi   = lane >> 4;                  // 0 or 1

  const float* xrow = x + (size_t)(row0 + m) * NFEAT;
  const float* wcol = W + col0 + m;

  v8f c = {0.f, 0.f, 0.f, 0.f, 0.f, 0.f, 0.f, 0.f};
#pragma unroll
  for (int k = 0; k < NFEAT; k += 4) {
    v2f a = *(const v2f*)(xrow + k + 2 * hi);            // x[row0+m][k+2hi .. +1]
    v2f b;
    b.x = wcol[(size_t)(k + 2 * hi)     * NHID];          // W[k+2hi  ][col0+m]
    b.y = wcol[(size_t)(k + 2 * hi + 1) * NHID];          // W[k+2hi+1][col0+m]
    c = __builtin_amdgcn_wmma_f32_16x16x4_f32(
        /*neg_a=*/false, a, /*neg_b=*/false, b,
        /*c_mod=*/(short)0, c, /*reuse_a=*/false, /*reuse_b=*/false);
  }
#pragma unroll
  for (int r = 0; r < 8; ++r) {
    int ro = row0 + r + 8 * hi;                           // D layout row
    h[(size_t)ro * NHID + col0 + m] = c[r];
  }
}

// ---------------------------------------------------------------------------
// Kernel 2: per-node attention projections  s1[i] = h[i]·a1, s2[i] = h[i]·a2
// One wave per node; lane holds one float4 (128/32 = 4 elems); shfl reduce.
// ---------------------------------------------------------------------------
__global__ __launch_bounds__(256) void gat_scores(
    const float* __restrict__ h, const float* __restrict__ a,
    float* __restrict__ s1, float* __restrict__ s2) {
  const int wid  = (blockIdx.x * blockDim.x + threadIdx.x) >> 5;
  const int lane = threadIdx.x & 31;
  if (wid >= NNODES) return;

  float4 hv  = ((const float4*)(h + (size_t)wid * NHID))[lane];
  float4 a1v = ((const float4*)(a))[lane];          // a[0..127]
  float4 a2v = ((const float4*)(a + NHID))[lane];   // a[128..255]

  float d1 = hv.x * a1v.x + hv.y * a1v.y + hv.z * a1v.z + hv.w * a1v.w;
  float d2 = hv.x * a2v.x + hv.y * a2v.y + hv.z * a2v.z + hv.w * a2v.w;
#pragma unroll
  for (int off = 16; off > 0; off >>= 1) {
    d1 += __shfl_down(d1, off, 32);
    d2 += __shfl_down(d2, off, 32);
  }
  if (lane == 0) { s1[wid] = d1; s2[wid] = d2; }
}

// ---------------------------------------------------------------------------
// Kernel 3: per-edge scatter. One wave per edge.
//   w = exp(-leaky_relu(s1[src]+s2[dst], 0.2))
//   rowsum[src]  += w                (lane 0)
//   hprime[src]  += w * h[dst]       (128 f32 -> 4 global_atomic_add_f32/lane)
// h and hprime are L2-resident (25.6 MB each, 192 MB L2).
// ---------------------------------------------------------------------------
__global__ __launch_bounds__(256) void gat_edges(
    const int* __restrict__ src, const int* __restrict__ dst,
    const float* __restrict__ s1, const float* __restrict__ s2,
    const float* __restrict__ h,
    float* __restrict__ rowsum, float* __restrict__ hprime) {
  const int eid  = (blockIdx.x * blockDim.x + threadIdx.x) >> 5;
  const int lane = threadIdx.x & 31;
  if (eid >= NEDGES) return;

  const int s = src[eid];
  const int d = dst[eid];
  float sc = s1[s] + s2[d];
  float lr = sc > 0.f ? sc : ALPHA * sc;
  float w  = __expf(-lr);

  if (lane == 0) unsafeAtomicAdd(rowsum + s, w);

  float4 hv = ((const float4*)(h + (size_t)d * NHID))[lane];
  float* out = hprime + (size_t)s * NHID + lane * 4;
  unsafeAtomicAdd(out + 0, w * hv.x);
  unsafeAtomicAdd(out + 1, w * hv.y);
  unsafeAtomicAdd(out + 2, w * hv.z);
  unsafeAtomicAdd(out + 3, w * hv.w);
}

// ---------------------------------------------------------------------------
// Kernel 4: out = elu(h - hprime / (rowsum + 1e-16))
// ---------------------------------------------------------------------------
__global__ __launch_bounds__(256) void gat_finalize(
    const float* __restrict__ h, const float* __restrict__ hprime,
    const float* __restrict__ rowsum, float* __restrict__ out) {
  const int i = blockIdx.x * blockDim.x + threadIdx.x;
  if (i >= NNODES * NHID) return;
  const int row = i >> 7;
  float nb = hprime[i] / (rowsum[row] + 1e-16f);
  float z  = h[i] - nb;
  out[i]   = z > 0.f ? z : expm1f(z);
}

extern "C" void kernel_launch(void* const* d_in, const int* in_sizes, int n_in,
                              void* d_out, int out_size, void* d_ws, size_t ws_size,
                              hipStream_t stream) {
  const float* x  = (const float*)d_in[0];          // N x 128
  const float* W  = (const float*)d_in[1];          // 128 x 128
  const float* a  = (const float*)d_in[2];          // 1 x 256
  const int*   ei = (const int*)d_in[3];            // 2 x E (src row then dst row)
  const int* src = ei;
  const int* dst = ei + NEDGES;

  // Workspace layout (floats)
  float* h      = (float*)d_ws;                                 // N*128
  float* hprime = h + (size_t)NNODES * NHID;                    // N*128
  float* s1     = hprime + (size_t)NNODES * NHID;               // N
  float* s2     = s1 + NNODES;                                  // N
  float* rowsum = s2 + NNODES;                                  // N
  float* out    = (float*)d_out;

  // 1) h = x @ W   (N/16 = 3125 row-tiles, 8 column-tile waves per block)
  gat_gemm_wmma<<<NNODES / 16, 256, 0, stream>>>(x, W, h);

  // 2) s1, s2  (one wave per node, 8 waves per block)
  gat_scores<<<(NNODES + 7) / 8, 256, 0, stream>>>(h, a, s1, s2);

  // 3) zero accumulators (graph-capturable async memsets)
  hipMemsetAsync(hprime, 0, (size_t)NNODES * NHID * sizeof(float), stream);
  hipMemsetAsync(rowsum, 0, (size_t)NNODES * sizeof(float), stream);

  // 4) edge scatter (one wave per edge, 8 edges per block)
  gat_edges<<<NEDGES / 8, 256, 0, stream>>>(src, dst, s1, s2, h, rowsum, hprime);

  // 5) finalize
  gat_finalize<<<(NNODES * NHID + 255) / 256, 256, 0, stream>>>(h, hprime, rowsum, out);
}